// ContrastiveGNN_13520557048098
// MI455X (gfx1250) — compile-verified
//
#include <hip/hip_runtime.h>

// ---------------------------------------------------------------------------
// ContrastiveGNN for MI455X (gfx1250): 3x GCNConv + projector MLP.
// GEMMs use V_WMMA_F32_16X16X4_F32 (exact f32). Edge aggregation is a pure
// gather + global_atomic_add_f32 pass (norms folded into per-node prescale).
// ---------------------------------------------------------------------------

typedef __attribute__((ext_vector_type(2))) float v2f;
typedef __attribute__((ext_vector_type(8))) float v8f;

// ----------------------------- utility kernels -----------------------------

__global__ void deg_init_kernel(float* __restrict__ deg, int n) {
  int i = blockIdx.x * blockDim.x + threadIdx.x;
  if (i < n) deg[i] = 1.0f;  // self-loop
}

__global__ void deg_count_kernel(const long long* __restrict__ ei,
                                 float* __restrict__ deg, int E) {
  int e = blockIdx.x * blockDim.x + threadIdx.x;
  if (e < E) unsafeAtomicAdd(&deg[(int)ei[(size_t)E + e]], 1.0f);
}

__global__ void deg_rsqrt_kernel(const float* __restrict__ deg,
                                 float* __restrict__ rs, int n) {
  int i = blockIdx.x * blockDim.x + threadIdx.x;
  if (i < n) rs[i] = rsqrtf(deg[i]);
}

__global__ void zero_kernel(float4* __restrict__ p, int n4) {
  int i = blockIdx.x * blockDim.x + threadIdx.x;
  if (i < n4) p[i] = float4{0.0f, 0.0f, 0.0f, 0.0f};
}

// ------------------------------ WMMA GEMM ----------------------------------
// One wave computes a 16x64 output tile of out = A[M,K] @ W[K,64].
// MODE 0 (GCN):   out = acc * rs[row]   (prescaled h; agg zeroed elsewhere)
// MODE 1 (LIN):   out = acc + bias
// MODE 2 (LINR):  out = relu(acc + bias)
//
// LDS layout interleaves k-row pairs: element (k,n) -> sW[(k>>1)*PSTRIDE+2n+(k&1)]
// so each B fragment is a single aligned ds_load_b64 with no register shuffles;
// PSTRIDE % 64 == 32 makes the two half-wave bank sets disjoint.
template <int K, int MODE>
__global__ __launch_bounds__(256) void gemm_wmma(
    const float* __restrict__ A, const float* __restrict__ W,
    const float* __restrict__ bias, const float* __restrict__ rs,
    float* __restrict__ out, int M) {
  constexpr int CH = 64;        // K-chunk staged in LDS
  constexpr int PSTRIDE = 160;  // words per interleaved pair-row (128 + 32 pad)
  __shared__ float sW[(CH / 2) * PSTRIDE];

  const int wave = blockIdx.x * 8 + ((int)threadIdx.x >> 5);
  const int lane = (int)threadIdx.x & 31;
  const int mBlocks = M >> 4;                // M is a multiple of 16 (100000)
  const bool active = wave < mBlocks;
  const int m0 = active ? (wave << 4) : 0;   // inactive waves shadow block 0
  const int half = lane >> 4;
  const int l = lane & 15;

  const float* arow = A + (size_t)(m0 + l) * K + 2 * half;

  v8f c0 = {}, c1 = {}, c2 = {}, c3 = {};

  for (int kc = 0; kc < K; kc += CH) {
    __syncthreads();  // all waves alive: inactive ones shadow-compute
    for (int i = threadIdx.x; i < CH * 64; i += 256) {
      const int k = i >> 6, n = i & 63;
      sW[(k >> 1) * PSTRIDE + (n << 1) + (k & 1)] = W[(size_t)(kc + k) * 64 + n];
    }
    __syncthreads();

#pragma unroll
    for (int kk = 0; kk < CH; kk += 4) {
      v2f a;
      a.x = arow[kc + kk];
      a.y = arow[kc + kk + 1];
      // pair-row index: (kk + 2*half)/2 ; kk is even so this is exact
      const float* bp = &sW[((kk >> 1) + half) * PSTRIDE + (l << 1)];
      const v2f b0 = *(const v2f*)(bp);
      const v2f b1 = *(const v2f*)(bp + 32);
      const v2f b2 = *(const v2f*)(bp + 64);
      const v2f b3 = *(const v2f*)(bp + 96);
      c0 = __builtin_amdgcn_wmma_f32_16x16x4_f32(false, a, false, b0, (short)0, c0, false, false);
      c1 = __builtin_amdgcn_wmma_f32_16x16x4_f32(false, a, false, b1, (short)0, c1, false, false);
      c2 = __builtin_amdgcn_wmma_f32_16x16x4_f32(false, a, false, b2, (short)0, c2, false, false);
      c3 = __builtin_amdgcn_wmma_f32_16x16x4_f32(false, a, false, b3, (short)0, c3, false, false);
    }
  }

  if (!active) return;

#pragma unroll
  for (int r = 0; r < 8; ++r) {
    const int row = m0 + r + 8 * half;
    const size_t base = (size_t)row * 64 + l;
    if (MODE == 0) {
      const float s = rs[row];
      out[base + 0]  = c0[r] * s;
      out[base + 16] = c1[r] * s;
      out[base + 32] = c2[r] * s;
      out[base + 48] = c3[r] * s;
    } else {
      float v0 = c0[r] + bias[l];
      float v1 = c1[r] + bias[16 + l];
      float v2 = c2[r] + bias[32 + l];
      float v3 = c3[r] + bias[48 + l];
      if (MODE == 2) {
        v0 = fmaxf(v0, 0.0f); v1 = fmaxf(v1, 0.0f);
        v2 = fmaxf(v2, 0.0f); v3 = fmaxf(v3, 0.0f);
      }
      out[base + 0]  = v0;
      out[base + 16] = v1;
      out[base + 32] = v2;
      out[base + 48] = v3;
    }
  }
}

// --------------------------- edge aggregation ------------------------------
// agg[dst] += hs[src]; 16 lanes per edge, float4 per lane, f32 HW atomics.
__global__ void edge_agg_kernel(const long long* __restrict__ ei,
                                const float* __restrict__ hs,
                                float* __restrict__ agg, int E) {
  const int gid = blockIdx.x * blockDim.x + threadIdx.x;
  const int e = gid >> 4;
  if (e >= E) return;
  const int f = (gid & 15) << 2;
  const int src = (int)ei[e];
  const int dst = (int)ei[(size_t)E + e];
  const float4 h4 = *reinterpret_cast<const float4*>(hs + (size_t)src * 64 + f);
  float* ap = agg + (size_t)dst * 64 + f;
  unsafeAtomicAdd(ap + 0, h4.x);
  unsafeAtomicAdd(ap + 1, h4.y);
  unsafeAtomicAdd(ap + 2, h4.z);
  unsafeAtomicAdd(ap + 3, h4.w);
}

// ------------------------------ finalize -----------------------------------
// act = relu((agg + hs) * rs[node] + b[f])
__global__ void gcn_finalize_kernel(const float* __restrict__ agg,
                                    const float* __restrict__ hs,
                                    const float* __restrict__ rs,
                                    const float* __restrict__ bias,
                                    float* __restrict__ act, int total) {
  const int i = blockIdx.x * blockDim.x + threadIdx.x;
  if (i >= total) return;
  const float v = (agg[i] + hs[i]) * rs[i >> 6] + bias[i & 63];
  act[i] = fmaxf(v, 0.0f);
}

// ------------------------------ host entry ---------------------------------

extern "C" void kernel_launch(void* const* d_in, const int* in_sizes, int n_in,
                              void* d_out, int out_size, void* d_ws, size_t ws_size,
                              hipStream_t stream) {
  const float* x   = (const float*)d_in[0];
  const float* W1  = (const float*)d_in[1];
  const float* b1  = (const float*)d_in[2];
  const float* W2  = (const float*)d_in[3];
  const float* b2  = (const float*)d_in[4];
  const float* W3  = (const float*)d_in[5];
  const float* b3  = (const float*)d_in[6];
  const float* Pw1 = (const float*)d_in[7];
  const float* Pb1 = (const float*)d_in[8];
  const float* Pw2 = (const float*)d_in[9];
  const float* Pb2 = (const float*)d_in[10];
  const long long* ei = (const long long*)d_in[11];

  const int N = in_sizes[0] / 256;   // 100000
  const int E = in_sizes[11] / 2;    // 1600000

  // workspace layout (all re-initialized every call)
  char* ws = (char*)d_ws;
  float* deg = (float*)ws; ws += (size_t)N * 4;
  float* rs  = (float*)ws; ws += (size_t)N * 4;
  float* hs  = (float*)ws; ws += (size_t)N * 64 * 4;
  float* agg = (float*)ws; ws += (size_t)N * 64 * 4;
  float* act = (float*)ws; ws += (size_t)N * 64 * 4;
  float* z1  = hs;  // reuse hs buffer for projector hidden

  const dim3 blk(256);
  const int mBlocks  = (N + 15) / 16;
  const int gemmGrid = (mBlocks + 7) / 8;
  const int edgeGrid = (E * 16 + 255) / 256;
  const int elemGrid = (N * 64 + 255) / 256;
  const int zeroGrid = (N * 16 + 255) / 256;  // N*64/4 float4s
  const int nGrid    = (N + 255) / 256;
  const int eGrid    = (E + 255) / 256;

  // degree (A + I): deg = 1 + in-degree; rs = deg^{-1/2}
  deg_init_kernel<<<nGrid, blk, 0, stream>>>(deg, N);
  deg_count_kernel<<<eGrid, blk, 0, stream>>>(ei, deg, E);
  deg_rsqrt_kernel<<<nGrid, blk, 0, stream>>>(deg, rs, N);

  // GCN layer 1 (K=256)
  gemm_wmma<256, 0><<<gemmGrid, blk, 0, stream>>>(x, W1, b1, rs, hs, N);
  zero_kernel<<<zeroGrid, blk, 0, stream>>>((float4*)agg, N * 16);
  edge_agg_kernel<<<edgeGrid, blk, 0, stream>>>(ei, hs, agg, E);
  gcn_finalize_kernel<<<elemGrid, blk, 0, stream>>>(agg, hs, rs, b1, act, N * 64);

  // GCN layer 2 (K=64)
  gemm_wmma<64, 0><<<gemmGrid, blk, 0, stream>>>(act, W2, b2, rs, hs, N);
  zero_kernel<<<zeroGrid, blk, 0, stream>>>((float4*)agg, N * 16);
  edge_agg_kernel<<<edgeGrid, blk, 0, stream>>>(ei, hs, agg, E);
  gcn_finalize_kernel<<<elemGrid, blk, 0, stream>>>(agg, hs, rs, b2, act, N * 64);

  // GCN layer 3 (K=64)
  gemm_wmma<64, 0><<<gemmGrid, blk, 0, stream>>>(act, W3, b3, rs, hs, N);
  zero_kernel<<<zeroGrid, blk, 0, stream>>>((float4*)agg, N * 16);
  edge_agg_kernel<<<edgeGrid, blk, 0, stream>>>(ei, hs, agg, E);
  gcn_finalize_kernel<<<elemGrid, blk, 0, stream>>>(agg, hs, rs, b3, act, N * 64);

  // projector MLP: relu(act @ Pw1 + Pb1) @ Pw2 + Pb2
  gemm_wmma<64, 2><<<gemmGrid, blk, 0, stream>>>(act, Pw1, Pb1, rs, z1, N);
  gemm_wmma<64, 1><<<gemmGrid, blk, 0, stream>>>(z1, Pw2, Pb2, rs, (float*)d_out, N);
}